// Transducer_28656021799147
// MI455X (gfx1250) — compile-verified
//
#include <hip/hip_runtime.h>
#include <cmath>

// ---------------------------------------------------------------------------
// RNN-T loss on MI455X (gfx1250, wave32).
// All GEMM stages use V_WMMA_F32_16X16X4_F32 (full fp32).
// Joint log-softmax is refactored as  lse = Mf + Mg + log(Pf @ Pg^T)  so the
// (N,T,U+1,V) joint tensor is never materialized.
// LDS staging uses GLOBAL_LOAD_ASYNC_TO_LDS_B128 when the toolchain exposes
// the gfx1250 async-to-LDS builtins (falls back to plain staging otherwise).
// ---------------------------------------------------------------------------

typedef __attribute__((ext_vector_type(2))) float v2f;
typedef __attribute__((ext_vector_type(8))) float v8f;
typedef int v4i __attribute__((vector_size(4 * sizeof(int))));

#define DEV_INLINE __device__ __forceinline__

#if defined(__gfx1250__) && \
    __has_builtin(__builtin_amdgcn_global_load_async_to_lds_b128) && \
    __has_builtin(__builtin_amdgcn_s_wait_asynccnt)
#define USE_ASYNC_LDS 1
#else
#define USE_ASYNC_LDS 0
#endif

constexpr int kN = 4, kT = 512, kU = 128, kU1 = 129, kV = 256;
constexpr int kFEAT = 1024, kHID = 512, kG = 2048;

// Workspace layout (float offsets)
constexpr size_t OFF_FLOG  = 0;                         // feat_logits N*T*V
constexpr size_t OFF_MF    = OFF_FLOG  + (size_t)kN*kT*kV;      // N*T
constexpr size_t OFF_PF    = OFF_MF    + (size_t)kN*kT;         // N*T*V
constexpr size_t OFF_GLOG  = OFF_PF    + (size_t)kN*kT*kV;      // lm_out N*U1*V
constexpr size_t OFF_MG    = OFF_GLOG  + (size_t)kN*kU1*kV;     // pad 528
constexpr size_t OFF_PG    = OFF_MG    + 528;                   // N*U1*V
constexpr size_t OFF_HSEQ0 = OFF_PG    + (size_t)kN*kU1*kV;     // N*U1*HID
constexpr size_t OFF_HSEQ1 = OFF_HSEQ0 + (size_t)kN*kU1*kHID;   // N*U1*HID
constexpr size_t OFF_STATE = OFF_HSEQ1 + (size_t)kN*kU1*kHID;   // 12288
constexpr size_t OFF_BLANK = OFF_STATE + 12288;                 // N*T*U1
constexpr size_t OFF_EMIT  = OFF_BLANK + (size_t)kN*kT*kU1;     // N*T*U
constexpr size_t OFF_WT    = OFF_EMIT  + (size_t)kN*kT*kU;      // 4 * 2048*512

DEV_INLINE v8f wmma_f32_k4(v2f a, v2f b, v8f c) {
  // D(16x16,f32) = A(16x4,f32) * B(4x16,f32) + C
  return __builtin_amdgcn_wmma_f32_16x16x4_f32(false, a, false, b,
                                               (short)0, c, false, false);
}

DEV_INLINE float sigmoidf(float x) { return 1.0f / (1.0f + expf(-x)); }

DEV_INLINE float lse2(float a, float b) {
  float m = fmaxf(a, b);
  if (m == -INFINITY) return -INFINITY;
  return m + log1pf(expf(-fabsf(a - b)));
}

#if USE_ASYNC_LDS
// 16-byte async global->LDS copy (per lane), tracked by ASYNCcnt.
DEV_INLINE void async_copy16(void* lds_dst, const void* gsrc) {
  __builtin_amdgcn_global_load_async_to_lds_b128(
      (__attribute__((address_space(1))) v4i*)gsrc,
      (__attribute__((address_space(3))) v4i*)lds_dst, 0, 0);
}
DEV_INLINE void async_wait0() { __builtin_amdgcn_s_wait_asynccnt(0); }
#endif

// ---------------------------------------------------------------------------
__global__ void k_zero(float* p, int n) {
  int i = blockIdx.x * 256 + threadIdx.x;
  if (i < n) p[i] = 0.0f;
}

// in: rows x cols (row-major) -> out: cols x rows (row-major)
__global__ void k_transpose(const float* __restrict__ in, float* __restrict__ out,
                            int rows, int cols) {
  size_t i = (size_t)blockIdx.x * 256 + threadIdx.x;
  if (i < (size_t)rows * cols) {
    int r = (int)(i / cols);
    int c = (int)(i % cols);
    out[(size_t)c * rows + r] = in[i];
  }
}

// ---------------------------------------------------------------------------
// C = A(MxK) @ B(Kx256) + bias; also rowmax and exp(C - rowmax) per row.
// Block: 256 thr (8 waves), each block = 16 rows x 256 cols; wave w covers
// column tiles [32w,32w+16) and [32w+16,32w+32).
__global__ __launch_bounds__(256) void k_gemm_maxexp(
    const float* __restrict__ A, const float* __restrict__ B,
    const float* __restrict__ bias, float* __restrict__ logits,
    float* __restrict__ rowmax, float* __restrict__ rowexp, int K, int Mvalid) {
  __shared__ __align__(16) float As[16 * 128];
  __shared__ float Ls[16 * kV];
  __shared__ float Red[16 * 16];
  __shared__ float Mv[16];

  const int tid = threadIdx.x, lane = tid & 31, w = tid >> 5;
  const int m0 = blockIdx.x * 16;
  const int row = lane & 15, kh = lane >> 4, kb = 2 * kh, col = lane & 15;
  const int c0 = w * 32;

  v8f acc0 = {0,0,0,0,0,0,0,0};
  v8f acc1 = {0,0,0,0,0,0,0,0};

  for (int k0 = 0; k0 < K; k0 += 128) {
#if USE_ASYNC_LDS
    for (int i = 0; i < 2; i++) {   // 2 x 16B per thread
      int idx = (tid + i * 256) * 4;
      int r = idx >> 7, kk = idx & 127;
      int gm = m0 + r;
      if (gm < Mvalid) {
        async_copy16(&As[idx], A + (size_t)gm * K + k0 + kk);
      } else {
        As[idx] = 0.0f; As[idx + 1] = 0.0f; As[idx + 2] = 0.0f; As[idx + 3] = 0.0f;
      }
    }
    async_wait0();
#else
    for (int i = 0; i < 8; i++) {  // stage 16x128 A chunk, coalesced
      int idx = tid + i * 256;
      int r = idx >> 7, kk = idx & 127;
      int gm = m0 + r;
      As[idx] = (gm < Mvalid) ? A[(size_t)gm * K + k0 + kk] : 0.0f;
    }
#endif
    __syncthreads();
    for (int kk = 0; kk < 128; kk += 4) {
      v2f a;
      a.x = As[row * 128 + kk + kb];
      a.y = As[row * 128 + kk + kb + 1];
      const float* Bp = B + (size_t)(k0 + kk + kb) * kV;
      v2f b0; b0.x = Bp[c0 + col];        b0.y = Bp[kV + c0 + col];
      acc0 = wmma_f32_k4(a, b0, acc0);
      v2f b1; b1.x = Bp[c0 + 16 + col];   b1.y = Bp[kV + c0 + 16 + col];
      acc1 = wmma_f32_k4(a, b1, acc1);
    }
    __syncthreads();
  }

#pragma unroll
  for (int r = 0; r < 8; r++) {
    int rc = r + 8 * kh;
    Ls[rc * kV + c0 + col]      = acc0[r] + bias[c0 + col];
    Ls[rc * kV + c0 + 16 + col] = acc1[r] + bias[c0 + 16 + col];
  }
  __syncthreads();
  {
    int rr = tid >> 4, p = tid & 15;
    float m = -INFINITY;
    for (int j = 0; j < 16; j++) m = fmaxf(m, Ls[rr * kV + p * 16 + j]);
    Red[rr * 16 + p] = m;
  }
  __syncthreads();
  if (tid < 16) {
    float m = -INFINITY;
    for (int p = 0; p < 16; p++) m = fmaxf(m, Red[tid * 16 + p]);
    Mv[tid] = m;
    if (m0 + tid < Mvalid) rowmax[m0 + tid] = m;
  }
  __syncthreads();
  {
    int rr = tid >> 4, p = tid & 15;
    int gm = m0 + rr;
    if (gm < Mvalid) {
      for (int j = 0; j < 16; j++) {
        int cc = p * 16 + j;
        float v = Ls[rr * kV + cc];
        logits[(size_t)gm * kV + cc] = v;
        rowexp[(size_t)gm * kV + cc] = expf(v - Mv[rr]);
      }
    }
  }
}

// ---------------------------------------------------------------------------
// One pipelined LSTM step: blocks 0-3 do layer0 step s, blocks 4-7 layer1
// step s-1. h state is double-buffered by step parity. Each block owns 128
// hidden units: wave w computes the (i,f,g,o) 16x16 gate tiles for its 16
// hidden units over K=512(x)+512(h).
__global__ __launch_bounds__(256) void k_lstm_step(
    int s, const float* __restrict__ emb, const int* __restrict__ targets,
    const float* __restrict__ WTih0, const float* __restrict__ WThh0,
    const float* __restrict__ bih0, const float* __restrict__ bhh0,
    const float* __restrict__ WTih1, const float* __restrict__ WThh1,
    const float* __restrict__ bih1, const float* __restrict__ bhh1,
    float* __restrict__ hseq0, float* __restrict__ hseq1,
    float* __restrict__ stateZ) {
  const int L = blockIdx.x >> 2, jb = blockIdx.x & 3;
  const int step = (L == 0) ? s : (s - 1);
  if (step < 0 || step > kU) return;  // steps 0..128

  const int parity = step & 1;
  float* hread  = stateZ + L * 2048 + (parity ? 8192 : 0);
  float* hwrite = stateZ + L * 2048 + (parity ? 0 : 8192);
  float* cst    = stateZ + 4096 + L * 2048;
  const float* WTih = L ? WTih1 : WTih0;
  const float* WThh = L ? WThh1 : WThh0;
  const float* bih  = L ? bih1 : bih0;
  const float* bhh  = L ? bhh1 : bhh0;
  float* hseq = L ? hseq1 : hseq0;

  __shared__ __align__(16) float xs[4 * 512];
  __shared__ __align__(16) float hsd[4 * 512];
  const int tid = threadIdx.x;
#if USE_ASYNC_LDS
  for (int i = 0; i < 2; i++) {   // 2 x 16B per thread per array
    int idx = (tid + i * 256) * 4;
    int n = idx >> 9, k = idx & 511;
    const float* src;
    if (L == 0) {
      int tok = (step == 0) ? 0 : targets[n * kU + step - 1];
      src = emb + (size_t)tok * 512 + k;
    } else {
      src = hseq0 + ((size_t)(n * kU1) + step) * 512 + k;
    }
    async_copy16(&xs[idx], src);
    async_copy16(&hsd[idx], hread + idx);
  }
  async_wait0();
#else
  for (int i = 0; i < 8; i++) {
    int idx = tid + i * 256;
    int n = idx >> 9, k = idx & 511;
    float xv;
    if (L == 0) {
      int tok = (step == 0) ? 0 : targets[n * kU + step - 1];
      xv = emb[(size_t)tok * 512 + k];
    } else {
      xv = hseq0[((size_t)(n * kU1) + step) * 512 + k];
    }
    xs[idx] = xv;
    hsd[idx] = hread[idx];
  }
#endif
  __syncthreads();

  const int lane = tid & 31, w = tid >> 5;
  const int row = lane & 15, kh = lane >> 4, kb = 2 * kh, col = lane & 15;
  const int jw = jb * 128 + w * 16;

  v8f zero8 = {0,0,0,0,0,0,0,0};
  v8f acc[4] = {zero8, zero8, zero8, zero8};

  for (int k = 0; k < 512; k += 4) {  // x @ Wih^T
    int kk = k + kb;
    v2f a;
    a.x = (row < kN) ? xs[row * 512 + kk] : 0.0f;
    a.y = (row < kN) ? xs[row * 512 + kk + 1] : 0.0f;
#pragma unroll
    for (int gi = 0; gi < 4; gi++) {
      int bc = gi * 512 + jw + col;
      v2f b;
      b.x = WTih[(size_t)kk * kG + bc];
      b.y = WTih[(size_t)(kk + 1) * kG + bc];
      acc[gi] = wmma_f32_k4(a, b, acc[gi]);
    }
  }
  for (int k = 0; k < 512; k += 4) {  // h @ Whh^T
    int kk = k + kb;
    v2f a;
    a.x = (row < kN) ? hsd[row * 512 + kk] : 0.0f;
    a.y = (row < kN) ? hsd[row * 512 + kk + 1] : 0.0f;
#pragma unroll
    for (int gi = 0; gi < 4; gi++) {
      int bc = gi * 512 + jw + col;
      v2f b;
      b.x = WThh[(size_t)kk * kG + bc];
      b.y = WThh[(size_t)(kk + 1) * kG + bc];
      acc[gi] = wmma_f32_k4(a, b, acc[gi]);
    }
  }

  if (kh == 0) {  // rows 0..3 of each D tile = the 4 batch elements
#pragma unroll
    for (int r = 0; r < kN; r++) {
      int n = r, hj = jw + col;
      float gi_ = acc[0][r] + bih[hj] + bhh[hj];
      float gf  = acc[1][r] + bih[512 + hj] + bhh[512 + hj];
      float gg  = acc[2][r] + bih[1024 + hj] + bhh[1024 + hj];
      float go  = acc[3][r] + bih[1536 + hj] + bhh[1536 + hj];
      float cOld = cst[n * 512 + hj];
      float cN = sigmoidf(gf) * cOld + sigmoidf(gi_) * tanhf(gg);
      float hN = sigmoidf(go) * tanhf(cN);
      cst[n * 512 + hj] = cN;
      hwrite[n * 512 + hj] = hN;
      hseq[((size_t)(n * kU1) + step) * 512 + hj] = hN;
    }
  }
}

// ---------------------------------------------------------------------------
// S = Pf(T x V) @ Pg^T(V x U1) per batch; lse = log S + Mf + Mg;
// blank/emit = (f + g - lse) gathers. Block = 16 t-rows x 129 u-cols (9 waves).
__global__ __launch_bounds__(288) void k_joint(
    const float* __restrict__ Pf, const float* __restrict__ Mf,
    const float* __restrict__ Pg, const float* __restrict__ Mg,
    const float* __restrict__ flog, const float* __restrict__ glog,
    const int* __restrict__ targets, float* __restrict__ blank,
    float* __restrict__ emit) {
  const int n = blockIdx.x >> 5, rt = blockIdx.x & 31;
  const int t0 = rt * 16;
  const int tid = threadIdx.x, lane = tid & 31, w = tid >> 5;
  const int u0 = w * 16;
  const int row = lane & 15, kh = lane >> 4, kb = 2 * kh, col = lane & 15;
  const int u = u0 + col;
  const int uc = (u < kU1) ? u : (kU1 - 1);
  const float msk = (u < kU1) ? 1.0f : 0.0f;

  const float* Ap = Pf + ((size_t)(n * kT + t0 + row)) * kV + kb;
  const float* Bp = Pg + ((size_t)(n * kU1 + uc)) * kV + kb;

  v8f acc = {0,0,0,0,0,0,0,0};
  for (int k = 0; k < kV; k += 4) {
    v2f a; a.x = Ap[k]; a.y = Ap[k + 1];
    v2f b; b.x = Bp[k] * msk; b.y = Bp[k + 1] * msk;
    acc = wmma_f32_k4(a, b, acc);
  }

  if (u < kU1) {
    float mg = Mg[n * kU1 + u];
    float g0 = glog[(size_t)(n * kU1 + u) * kV];
    int tgt = (u < kU) ? targets[n * kU + u] : 0;
    float gt = (u < kU) ? glog[(size_t)(n * kU1 + u) * kV + tgt] : 0.0f;
#pragma unroll
    for (int r = 0; r < 8; r++) {
      int t = t0 + r + 8 * kh;
      float lse = logf(acc[r]) + Mf[n * kT + t] + mg;
      float f0 = flog[(size_t)(n * kT + t) * kV];
      blank[((size_t)(n * kT) + t) * kU1 + u] = f0 + g0 - lse;
      if (u < kU) {
        float ft = flog[(size_t)(n * kT + t) * kV + tgt];
        emit[((size_t)(n * kT) + t) * kU + u] = ft + gt - lse;
      }
    }
  }
}

// ---------------------------------------------------------------------------
// RNNT alpha DP: one wave per batch element; per t, an add prefix-scan for
// the emit cumsum and an LSE prefix-scan for cumlogsumexp (wave32 shuffles).
__global__ __launch_bounds__(128) void k_rnnt_dp(
    const float* __restrict__ blank, const float* __restrict__ emit,
    const int* __restrict__ input_lengths, const int* __restrict__ target_lengths,
    float* __restrict__ out) {
  __shared__ float alpha[kN][kU1 + 3];
  __shared__ float cbuf[kN][kU1 + 3];
  __shared__ float llv[kN];
  const int tid = threadIdx.x, lane = tid & 31, n = tid >> 5;
  const int ti = input_lengths[n] - 1;
  const int ui = target_lengths[n];

  // t = 0: alpha[u] = exclusive cumsum of emit[n,0,:]
  {
    const float* erow = emit + (size_t)(n * kT) * kU;
    if (lane == 0) cbuf[n][0] = 0.0f;
    float cy = 0.0f;
    for (int ch = 0; ch < 4; ch++) {
      float x = erow[ch * 32 + lane];
      for (int d = 1; d < 32; d <<= 1) {
        float o = __shfl_up(x, d, 32);
        if (lane >= d) x += o;
      }
      x += cy;
      cbuf[n][ch * 32 + lane + 1] = x;
      cy = __shfl(x, 31, 32);
    }
  }
  __syncthreads();
  for (int ch = 0; ch < 5; ch++) {
    int uu = ch * 32 + lane;
    if (uu < kU1) alpha[n][uu] = cbuf[n][uu];
  }
  __syncthreads();
  if (ti == 0 && lane == 0)
    llv[n] = alpha[n][ui] + blank[(size_t)(n * kT) * kU1 + ui];

  for (int t = 1; t < kT; t++) {
    __syncthreads();
    {  // cbuf = exclusive cumsum of emit[n,t,:]
      const float* erow = emit + (size_t)(n * kT + t) * kU;
      if (lane == 0) cbuf[n][0] = 0.0f;
      float cy = 0.0f;
      for (int ch = 0; ch < 4; ch++) {
        float x = erow[ch * 32 + lane];
        for (int d = 1; d < 32; d <<= 1) {
          float o = __shfl_up(x, d, 32);
          if (lane >= d) x += o;
        }
        x += cy;
        cbuf[n][ch * 32 + lane + 1] = x;
        cy = __shfl(x, 31, 32);
      }
    }
    __syncthreads();
    {  // alpha = cbuf + cumLSE(alpha + blank[t-1] - cbuf)
      const float* brow = blank + ((size_t)(n * kT) + (t - 1)) * kU1;
      float cyl = -INFINITY;
      for (int ch = 0; ch < 4; ch++) {
        int uu = ch * 32 + lane;
        float c = cbuf[n][uu];
        float x = alpha[n][uu] + brow[uu] - c;
        for (int d = 1; d < 32; d <<= 1) {
          float o = __shfl_up(x, d, 32);
          if (lane >= d) x = lse2(x, o);
        }
        x = lse2(x, cyl);
        alpha[n][uu] = c + x;
        cyl = __shfl(x, 31, 32);
      }
      if (lane == 0) {  // u = 128 tail
        float c = cbuf[n][kU];
        float x = lse2(alpha[n][kU] + brow[kU] - c, cyl);
        alpha[n][kU] = c + x;
      }
    }
    __syncthreads();
    if (t == ti && lane == 0)
      llv[n] = alpha[n][ui] + blank[((size_t)(n * kT) + ti) * kU1 + ui];
  }
  __syncthreads();
  if (tid == 0) out[0] = -(llv[0] + llv[1] + llv[2] + llv[3]) / (float)kN;
}

// ---------------------------------------------------------------------------
extern "C" void kernel_launch(void* const* d_in, const int* in_sizes, int n_in,
                              void* d_out, int out_size, void* d_ws, size_t ws_size,
                              hipStream_t stream) {
  const float* features = (const float*)d_in[0];
  const float* W_cls    = (const float*)d_in[1];
  const float* b_cls    = (const float*)d_in[2];
  const float* emb      = (const float*)d_in[3];
  const float* Wih0     = (const float*)d_in[4];
  const float* Whh0     = (const float*)d_in[5];
  const float* bih0     = (const float*)d_in[6];
  const float* bhh0     = (const float*)d_in[7];
  const float* Wih1     = (const float*)d_in[8];
  const float* Whh1     = (const float*)d_in[9];
  const float* bih1     = (const float*)d_in[10];
  const float* bhh1     = (const float*)d_in[11];
  const float* W_out    = (const float*)d_in[12];
  const float* b_out    = (const float*)d_in[13];
  const int* targets        = (const int*)d_in[14];
  const int* input_lengths  = (const int*)d_in[15];
  const int* target_lengths = (const int*)d_in[16];

  float* ws = (float*)d_ws;
  float* flog   = ws + OFF_FLOG;
  float* Mf     = ws + OFF_MF;
  float* Pf     = ws + OFF_PF;
  float* glog   = ws + OFF_GLOG;
  float* Mg     = ws + OFF_MG;
  float* Pg     = ws + OFF_PG;
  float* hseq0  = ws + OFF_HSEQ0;
  float* hseq1  = ws + OFF_HSEQ1;
  float* stateZ = ws + OFF_STATE;
  float* blank  = ws + OFF_BLANK;
  float* emit   = ws + OFF_EMIT;
  float* WT0i = ws + OFF_WT;
  float* WT0h = WT0i + (size_t)kG * kHID;
  float* WT1i = WT0h + (size_t)kG * kHID;
  float* WT1h = WT1i + (size_t)kG * kHID;

  // zero h (parity 0) and c state for both layers
  k_zero<<<32, 256, 0, stream>>>(stateZ, 8192);

  // pre-transpose LSTM weights: (2048,512) -> (512,2048), coalesced B loads
  k_transpose<<<4096, 256, 0, stream>>>(Wih0, WT0i, kG, kHID);
  k_transpose<<<4096, 256, 0, stream>>>(Whh0, WT0h, kG, kHID);
  k_transpose<<<4096, 256, 0, stream>>>(Wih1, WT1i, kG, kHID);
  k_transpose<<<4096, 256, 0, stream>>>(Whh1, WT1h, kG, kHID);

  // acoustic classifier: (N*T, FEAT) @ (FEAT, V) + max/exp
  k_gemm_maxexp<<<(kN * kT) / 16, 256, 0, stream>>>(
      features, W_cls, b_cls, flog, Mf, Pf, kFEAT, kN * kT);

  // 2-layer LSTM, software-pipelined (layer1 lags one step)
  for (int s = 0; s <= kU1; s++) {
    k_lstm_step<<<8, 256, 0, stream>>>(s, emb, targets, WT0i, WT0h, bih0, bhh0,
                                       WT1i, WT1h, bih1, bhh1, hseq0, hseq1,
                                       stateZ);
  }

  // LM projection: (N*U1, HID) @ (HID, V) + max/exp
  k_gemm_maxexp<<<(kN * kU1 + 15) / 16, 256, 0, stream>>>(
      hseq1, W_out, b_out, glog, Mg, Pg, kHID, kN * kU1);

  // joint lse GEMM -> blank/emit
  k_joint<<<kN * (kT / 16), 288, 0, stream>>>(Pf, Mf, Pg, Mg, flog, glog,
                                              targets, blank, emit);

  // alpha DP -> scalar loss
  k_rnnt_dp<<<1, 128, 0, stream>>>(blank, emit, input_lengths, target_lengths,
                                   (float*)d_out);
}